// FedUnlMlp_30434138259881
// MI455X (gfx1250) — compile-verified
//
#include <hip/hip_runtime.h>
#include <math.h>

// Problem constants (match reference)
#define BSEG   1024
#define HDIM   128
#define NITEM  100000
#define TLEN   51200
#define NTILES (NITEM / 16)   // 6250, exact

typedef __attribute__((ext_vector_type(2))) float v2f;
typedef __attribute__((ext_vector_type(8))) float v8f;

// ---------------------------------------------------------------------------
// Kernel 1: segment mean-pool of three tables -> user[b][h], averaged /3.
// seg arrays are sorted, so each block binary-searches its [lo,hi) range.
// grid = BSEG blocks, block = HDIM threads (4 waves).
// ---------------------------------------------------------------------------
__device__ __forceinline__ int lower_bound_i32(const int* __restrict__ a, int n, int key) {
    int lo = 0, hi = n;
    while (lo < hi) {
        int mid = (lo + hi) >> 1;
        if (a[mid] < key) lo = mid + 1; else hi = mid;
    }
    return lo;
}

__device__ __forceinline__ float seg_mean(const int* __restrict__ idx,
                                          const int* __restrict__ seg,
                                          const float* __restrict__ tab,
                                          int b, int h) {
    int lo = lower_bound_i32(seg, TLEN, b);
    int hi = lower_bound_i32(seg, TLEN, b + 1);
    float s = 0.f;
    for (int t = lo; t < hi; ++t)
        s += tab[(size_t)idx[t] * HDIM + h];
    return (hi > lo) ? s / (float)(hi - lo) : 0.f;
}

__global__ void __launch_bounds__(HDIM)
pool_kernel(const int* __restrict__ item_idx, const int* __restrict__ item_seg,
            const int* __restrict__ ent_idx,  const int* __restrict__ ent_seg,
            const int* __restrict__ word_idx, const int* __restrict__ word_seg,
            const float* __restrict__ item_tab, const float* __restrict__ ent_tab,
            const float* __restrict__ word_tab, float* __restrict__ user) {
    int b = blockIdx.x;
    int h = threadIdx.x;
    float acc = seg_mean(item_idx, item_seg, item_tab, b, h)
              + seg_mean(ent_idx,  ent_seg,  ent_tab,  b, h)
              + seg_mean(word_idx, word_seg, word_tab, b, h);
    user[b * HDIM + h] = acc * (1.0f / 3.0f);
}

// ---------------------------------------------------------------------------
// Kernel 2: lin = user(1024x128) @ W(128x100000) + bias, via V_WMMA_F32_16X16X4_F32.
// grid = (64 m-tiles, 32 n-groups), block = 256 (8 waves). Each wave owns one
// m-tile strip (A: 16x128 f32 in 32 v2f regs) and strides over n-tiles.
// A layout (ISA 32-bit A 16x4): lanes 0-15 -> K = 4k+{0,1}; lanes 16-31 -> K = 4k+{2,3}.
// B layout mirrors (row K striped): VGPR0 = K {0|2}, VGPR1 = K {1|3}, N = lane&15.
// C/D: VGPR r -> row m0 + r + 8*(lane>=16), col n0 + (lane&15).
// ---------------------------------------------------------------------------
__global__ void __launch_bounds__(256)
gemm_kernel(const float* __restrict__ user, const float* __restrict__ W,
            const float* __restrict__ bias, float* __restrict__ lin) {
    const int lane = threadIdx.x & 31;
    const int wave = threadIdx.x >> 5;
    const int m0   = blockIdx.x << 4;        // 0..1008
    const int lr   = lane & 15;
    const int hi   = lane >> 4;              // 0 or 1
    const int koff = hi << 1;                // 0 or 2

    // Pin A strip in registers: 16 rows x 128 K -> 32 x v2f per lane
    v2f a[32];
    const float* arow = user + (m0 + lr) * HDIM;
    #pragma unroll
    for (int k = 0; k < 32; ++k) {
        const float* ap = arow + 4 * k + koff;
        a[k] = v2f{ap[0], ap[1]};
    }

    const int nstream = (blockIdx.y << 3) + wave;     // 0..255
    for (int nt = nstream; nt < NTILES; nt += 256) {
        const int n0 = nt << 4;
        // Prefetch next n-tile's B column block (gfx1250 global_prefetch_b8)
        if (nt + 256 < NTILES)
            __builtin_prefetch(W + koff * NITEM + ((nt + 256) << 4) + lr, 0, 1);

        v8f c = {};
        #pragma unroll
        for (int k = 0; k < 32; ++k) {
            const float* bp = W + (size_t)(4 * k + koff) * NITEM + n0 + lr;
            v2f bb = v2f{bp[0], bp[NITEM]};
            c = __builtin_amdgcn_wmma_f32_16x16x4_f32(
                    false, a[k], false, bb, (short)0, c, false, false);
        }

        const float bi = bias[n0 + lr];
        const int rbase = m0 + (hi << 3);
        #pragma unroll
        for (int r = 0; r < 8; ++r)
            lin[(size_t)(rbase + r) * NITEM + n0 + lr] = c[r] + bi;
    }
}

// ---------------------------------------------------------------------------
// Kernel 3: per-row online softmax stats: row_m = max(lin), row_invS = 1/sum(exp(lin-m)).
// Note max(probs) == row_invS, which kernel 4 exploits.
// ---------------------------------------------------------------------------
__global__ void __launch_bounds__(256)
rowstats_kernel(const float* __restrict__ lin, float* __restrict__ row_m,
                float* __restrict__ row_invS) {
    __shared__ float sm[256], ss[256];
    const int b = blockIdx.x;
    const float* row = lin + (size_t)b * NITEM;
    float m = -INFINITY, s = 0.f;
    for (int i = threadIdx.x; i < NITEM; i += 256) {
        float x = row[i];
        if (x > m) { s = s * __expf(m - x) + 1.f; m = x; }
        else       { s += __expf(x - m); }
    }
    sm[threadIdx.x] = m; ss[threadIdx.x] = s;
    __syncthreads();
    for (int off = 128; off > 0; off >>= 1) {
        if (threadIdx.x < off) {
            float m1 = sm[threadIdx.x], s1 = ss[threadIdx.x];
            float m2 = sm[threadIdx.x + off], s2 = ss[threadIdx.x + off];
            if (m1 >= m2) { ss[threadIdx.x] = s1 + s2 * __expf(m2 - m1); }
            else          { ss[threadIdx.x] = s2 + s1 * __expf(m1 - m2); sm[threadIdx.x] = m2; }
        }
        __syncthreads();
    }
    if (threadIdx.x == 0) { row_m[b] = sm[0]; row_invS[b] = 1.f / ss[0]; }
}

// ---------------------------------------------------------------------------
// Kernel 4: in-place probs = exp(lin-m)*invS; fused accumulation of
// S2 = sum(exp(probs - m2)) with m2 = max(probs) = invS (no extra pass).
// ---------------------------------------------------------------------------
__global__ void __launch_bounds__(256)
probs_kernel(float* __restrict__ pl, const float* __restrict__ row_m,
             const float* __restrict__ row_invS, float* __restrict__ row_S2) {
    __shared__ float ss[256];
    const int b = blockIdx.x;
    const float m = row_m[b];
    const float invS = row_invS[b];
    const float m2 = invS;                  // max of probs
    float acc = 0.f;
    float* row = pl + (size_t)b * NITEM;
    for (int i = threadIdx.x; i < NITEM; i += 256) {
        float p = __expf(row[i] - m) * invS;
        row[i] = p;
        acc += __expf(p - m2);
    }
    ss[threadIdx.x] = acc;
    __syncthreads();
    for (int off = 128; off > 0; off >>= 1) {
        if (threadIdx.x < off) ss[threadIdx.x] += ss[threadIdx.x + off];
        __syncthreads();
    }
    if (threadIdx.x == 0) row_S2[b] = ss[0];
}

// ---------------------------------------------------------------------------
// Kernel 5: logp[b,label] = probs[b,label] - m2 - log(S2); loss = -mean.
// Also emits labels (as float) into d_out. Single block of BSEG threads.
// ---------------------------------------------------------------------------
__global__ void __launch_bounds__(BSEG)
loss_kernel(const float* __restrict__ probs, const int* __restrict__ labels,
            const float* __restrict__ row_invS, const float* __restrict__ row_S2,
            float* __restrict__ out_labels, float* __restrict__ out_loss) {
    __shared__ float ss[BSEG];
    const int b = threadIdx.x;
    const int lab = labels[b];
    const float p  = probs[(size_t)b * NITEM + lab];
    const float lp = p - row_invS[b] - logf(row_S2[b]);
    out_labels[b] = (float)lab;
    ss[b] = lp;
    __syncthreads();
    for (int off = BSEG / 2; off > 0; off >>= 1) {
        if (b < off) ss[b] += ss[b + off];
        __syncthreads();
    }
    if (b == 0) out_loss[0] = -ss[0] / (float)BSEG;
}

// ---------------------------------------------------------------------------
extern "C" void kernel_launch(void* const* d_in, const int* in_sizes, int n_in,
                              void* d_out, int out_size, void* d_ws, size_t ws_size,
                              hipStream_t stream) {
    const int*   item_idx  = (const int*)  d_in[0];
    const int*   item_seg  = (const int*)  d_in[1];
    const int*   ent_idx   = (const int*)  d_in[2];
    const int*   ent_seg   = (const int*)  d_in[3];
    const int*   word_idx  = (const int*)  d_in[4];
    const int*   word_seg  = (const int*)  d_in[5];
    const int*   labels    = (const int*)  d_in[6];
    const float* item_tab  = (const float*)d_in[7];
    const float* ent_tab   = (const float*)d_in[8];
    const float* word_tab  = (const float*)d_in[9];
    const float* rec_w     = (const float*)d_in[10];
    const float* rec_b     = (const float*)d_in[11];

    float* out      = (float*)d_out;
    float* probs    = out;                                   // B*N
    float* out_lab  = out + (size_t)BSEG * NITEM;            // B
    float* out_loss = out_lab + BSEG;                        // 1

    // workspace layout
    float* user     = (float*)d_ws;                          // B*H
    float* row_m    = user + BSEG * HDIM;                    // B
    float* row_invS = row_m + BSEG;                          // B
    float* row_S2   = row_invS + BSEG;                       // B

    pool_kernel<<<BSEG, HDIM, 0, stream>>>(item_idx, item_seg, ent_idx, ent_seg,
                                           word_idx, word_seg, item_tab, ent_tab,
                                           word_tab, user);
    gemm_kernel<<<dim3(64, 32), 256, 0, stream>>>(user, rec_w, rec_b, probs);
    rowstats_kernel<<<BSEG, 256, 0, stream>>>(probs, row_m, row_invS);
    probs_kernel<<<BSEG, 256, 0, stream>>>(probs, row_m, row_invS, row_S2);
    loss_kernel<<<1, BSEG, 0, stream>>>(probs, labels, row_invS, row_S2,
                                        out_lab, out_loss);
}